// GATRefiner_52596169507167
// MI455X (gfx1250) — compile-verified
//
#include <hip/hip_runtime.h>
#include <stdint.h>

// Problem constants (match reference setup_inputs()).
#define NN    50000       // nodes
#define EE    400000      // edges (before self loops)
#define INCH  128
#define OUTC  64
#define H1    8           // heads layer 1
#define NEG   0.2f

typedef float v2f __attribute__((ext_vector_type(2)));
typedef float v8f __attribute__((ext_vector_type(8)));

// ---------------------------------------------------------------------------
// util kernels
// ---------------------------------------------------------------------------
__global__ void fill_u32_kernel(uint32_t* p, uint32_t v, long long n) {
  long long i = (long long)blockIdx.x * blockDim.x + threadIdx.x;
  if (i < n) p[i] = v;
}

// ---------------------------------------------------------------------------
// fp32 WMMA GEMM: C[M,Nc] = A[M,K] * B[K,Nc]
// one wave -> 16 rows x 64 cols strip; K stepped by 4 via v_wmma_f32_16x16x4_f32
// Requires M%16==0, K%4==0, Nc%64==0 (true here: M=50000, K in {128,64},
// Nc in {512,64}).
// ---------------------------------------------------------------------------
__global__ void gemm_wmma_f32_kernel(const float* __restrict__ A,
                                     const float* __restrict__ B,
                                     float* __restrict__ C,
                                     int M, int K, int Nc) {
  const int wid  = (blockIdx.x * blockDim.x + threadIdx.x) >> 5;
  const int lane = threadIdx.x & 31;
  const int tilesN   = Nc >> 6;          // 64-col groups
  const int rowTiles = M >> 4;
  if (wid >= rowTiles * tilesN) return;  // whole wave exits together

  const int rt = wid / tilesN;
  const int cg = wid - rt * tilesN;
  const int m0 = rt << 4;
  const int n0 = cg << 6;
  const int ml = lane & 15;              // M index (A) / N index (B,C)
  const int kh = lane >> 4;              // K half-select

  v8f acc0 = {}, acc1 = {}, acc2 = {}, acc3 = {};
  const float* arow = A + (size_t)(m0 + ml) * K;

  for (int k = 0; k < K; k += 4) {
    const int ka = k + 2 * kh;           // lanes 0-15: k,k+1; lanes 16-31: k+2,k+3
    v2f a;
    a.x = arow[ka];
    a.y = arow[ka + 1];
    const float* b0p = B + (size_t)ka * Nc + n0 + ml;
    const float* b1p = b0p + Nc;
    v2f b0; b0.x = b0p[0];  b0.y = b1p[0];
    v2f b1; b1.x = b0p[16]; b1.y = b1p[16];
    v2f b2; b2.x = b0p[32]; b2.y = b1p[32];
    v2f b3; b3.x = b0p[48]; b3.y = b1p[48];
    acc0 = __builtin_amdgcn_wmma_f32_16x16x4_f32(false, a, false, b0, (short)0, acc0, false, false);
    acc1 = __builtin_amdgcn_wmma_f32_16x16x4_f32(false, a, false, b1, (short)0, acc1, false, false);
    acc2 = __builtin_amdgcn_wmma_f32_16x16x4_f32(false, a, false, b2, (short)0, acc2, false, false);
    acc3 = __builtin_amdgcn_wmma_f32_16x16x4_f32(false, a, false, b3, (short)0, acc3, false, false);
  }

  // C/D layout: VGPR v -> row m0 + v + 8*(lane>=16), col n0 + (lane&15) (+16t)
  const int mr = m0 + 8 * kh;
#pragma unroll
  for (int v = 0; v < 8; ++v) {
    float* crow = C + (size_t)(mr + v) * Nc + n0 + ml;
    crow[0]  = acc0[v];
    crow[16] = acc1[v];
    crow[32] = acc2[v];
    crow[48] = acc3[v];
  }
}

// ---------------------------------------------------------------------------
// attention scalars: a_s[n,h] = sum_c h[n,h,c]*att_src[h,c]; same for a_d.
// one wave per (node, head); 64 channels -> lane and lane+32.
// ---------------------------------------------------------------------------
__global__ void att_scores_kernel(const float* __restrict__ h,
                                  const float* __restrict__ att_src,
                                  const float* __restrict__ att_dst,
                                  float* __restrict__ a_s,
                                  float* __restrict__ a_d,
                                  int n_nodes, int heads) {
  const int wid  = (blockIdx.x * blockDim.x + threadIdx.x) >> 5;
  const int lane = threadIdx.x & 31;
  if (wid >= n_nodes * heads) return;
  const int n  = wid / heads;
  const int hd = wid - n * heads;
  const float* hrow = h + (size_t)n * heads * OUTC + hd * OUTC;
  const float* ws   = att_src + hd * OUTC;
  const float* wd   = att_dst + hd * OUTC;
  float vs = hrow[lane] * ws[lane] + hrow[lane + 32] * ws[lane + 32];
  float vd = hrow[lane] * wd[lane] + hrow[lane + 32] * wd[lane + 32];
#pragma unroll
  for (int off = 16; off; off >>= 1) {
    vs += __shfl_xor(vs, off, 32);
    vd += __shfl_xor(vd, off, 32);
  }
  if (lane == 0) {
    a_s[n * heads + hd] = vs;
    a_d[n * heads + hd] = vd;
  }
}

// ---------------------------------------------------------------------------
// edge helpers
// ---------------------------------------------------------------------------
__device__ __forceinline__ void edge_endpoints(const int* ei, int e, int& s, int& d) {
  if (e < EE) { s = ei[e]; d = ei[EE + e]; }
  else        { s = d = e - EE; }            // self loop
}

__device__ __forceinline__ float leaky(float x) {
  return x > 0.0f ? x : NEG * x;
}

__device__ __forceinline__ void atomicMaxF32(float* addr, float val) {
  if (val >= 0.0f) atomicMax((int*)addr, __float_as_int(val));
  else             atomicMin((unsigned int*)addr, __float_as_uint(val));
}

// pass A: segment max of edge scores into emax[dst,h]
__global__ void edge_max_kernel(const int* __restrict__ ei,
                                const float* __restrict__ a_s,
                                const float* __restrict__ a_d,
                                float* __restrict__ emax, int heads) {
  const long long tid = (long long)blockIdx.x * blockDim.x + threadIdx.x;
  const long long total = (long long)(EE + NN) * heads;
  if (tid >= total) return;
  const int e  = (int)(tid / heads);
  const int hd = (int)(tid - (long long)e * heads);
  int s, d; edge_endpoints(ei, e, s, d);
  const float ev = leaky(a_s[s * heads + hd] + a_d[d * heads + hd]);
  atomicMaxF32(&emax[d * heads + hd], ev);
}

// pass B: denom[dst,h] += exp(e - emax[dst,h])
__global__ void edge_expsum_kernel(const int* __restrict__ ei,
                                   const float* __restrict__ a_s,
                                   const float* __restrict__ a_d,
                                   const float* __restrict__ emax,
                                   float* __restrict__ denom, int heads) {
  const long long tid = (long long)blockIdx.x * blockDim.x + threadIdx.x;
  const long long total = (long long)(EE + NN) * heads;
  if (tid >= total) return;
  const int e  = (int)(tid / heads);
  const int hd = (int)(tid - (long long)e * heads);
  int s, d; edge_endpoints(ei, e, s, d);
  const float ev = leaky(a_s[s * heads + hd] + a_d[d * heads + hd]);
  atomicAdd(&denom[d * heads + hd], __expf(ev - emax[d * heads + hd]));
}

// pass C: acc[dst,c] += alpha * h[src,h,c]  (sum over heads folded in; one
// wave per (edge, head), 64 channels -> 2 atomics per lane)
__global__ void edge_aggregate_kernel(const int* __restrict__ ei,
                                      const float* __restrict__ a_s,
                                      const float* __restrict__ a_d,
                                      const float* __restrict__ emax,
                                      const float* __restrict__ denom,
                                      const float* __restrict__ h,
                                      float* __restrict__ acc, int heads) {
  const long long wid = ((long long)blockIdx.x * blockDim.x + threadIdx.x) >> 5;
  const int lane = threadIdx.x & 31;
  const long long total = (long long)(EE + NN) * heads;
  if (wid >= total) return;
  const int e  = (int)(wid / heads);
  const int hd = (int)(wid - (long long)e * heads);
  int s, d; edge_endpoints(ei, e, s, d);
  const float ev = leaky(a_s[s * heads + hd] + a_d[d * heads + hd]);
  const float alpha = __expf(ev - emax[d * heads + hd]) / denom[d * heads + hd];
  const float* hrow = h + (size_t)s * heads * OUTC + hd * OUTC;
  float* arow = acc + (size_t)d * OUTC;
  atomicAdd(&arow[lane],      alpha * hrow[lane]);
  atomicAdd(&arow[lane + 32], alpha * hrow[lane + 32]);
}

// finalize: out[n,c] = (acc[n,c] * invH + bias[c]) optionally ELU'd (in place OK)
__global__ void finalize_kernel(const float* __restrict__ acc,
                                const float* __restrict__ bias,
                                float* __restrict__ out,
                                float invH, int do_elu) {
  const long long tid = (long long)blockIdx.x * blockDim.x + threadIdx.x;
  if (tid >= (long long)NN * OUTC) return;
  const int c = (int)(tid & (OUTC - 1));
  float v = acc[tid] * invH + bias[c];
  if (do_elu) v = v > 0.0f ? v : (__expf(v) - 1.0f);
  out[tid] = v;
}

// ---------------------------------------------------------------------------
// host launcher
// ---------------------------------------------------------------------------
static inline unsigned cdiv(long long a, long long b) { return (unsigned)((a + b - 1) / b); }

extern "C" void kernel_launch(void* const* d_in, const int* in_sizes, int n_in,
                              void* d_out, int out_size, void* d_ws, size_t ws_size,
                              hipStream_t stream) {
  (void)in_sizes; (void)n_in; (void)out_size; (void)ws_size;
  const float* x    = (const float*)d_in[0];
  const int*   ei   = (const int*)d_in[1];
  const float* W1   = (const float*)d_in[2];
  const float* asw1 = (const float*)d_in[3];
  const float* adw1 = (const float*)d_in[4];
  const float* b1   = (const float*)d_in[5];
  const float* W2   = (const float*)d_in[6];
  const float* asw2 = (const float*)d_in[7];
  const float* adw2 = (const float*)d_in[8];
  const float* b2   = (const float*)d_in[9];
  float* out = (float*)d_out;

  // workspace carve-up (floats)
  float* ws   = (float*)d_ws;
  float* h1   = ws; ws += (size_t)NN * H1 * OUTC;   // 25.6M
  float* as1  = ws; ws += (size_t)NN * H1;
  float* ad1  = ws; ws += (size_t)NN * H1;
  float* em1  = ws; ws += (size_t)NN * H1;
  float* dn1  = ws; ws += (size_t)NN * H1;
  float* acc1 = ws; ws += (size_t)NN * OUTC;        // also h_mid after finalize
  float* h2   = ws; ws += (size_t)NN * OUTC;
  float* as2  = ws; ws += (size_t)NN;
  float* ad2  = ws; ws += (size_t)NN;
  float* em2  = ws; ws += (size_t)NN;
  float* dn2  = ws; ws += (size_t)NN;

  const int T = 256;
  const long long enh1 = (long long)(EE + NN) * H1;
  const long long enh2 = (long long)(EE + NN);
  const uint32_t NEG_INF = 0xFF800000u;

  // ---------------- layer 1 ----------------
  fill_u32_kernel<<<cdiv(NN * H1, T), T, 0, stream>>>((uint32_t*)em1, NEG_INF, NN * H1);
  fill_u32_kernel<<<cdiv(NN * H1, T), T, 0, stream>>>((uint32_t*)dn1, 0u, NN * H1);
  fill_u32_kernel<<<cdiv((long long)NN * OUTC, T), T, 0, stream>>>((uint32_t*)acc1, 0u, (long long)NN * OUTC);

  {  // h1 = x @ W1
    long long waves = ((long long)NN / 16) * (H1 * OUTC / 64);
    gemm_wmma_f32_kernel<<<cdiv(waves * 32, T), T, 0, stream>>>(x, W1, h1, NN, INCH, H1 * OUTC);
  }
  att_scores_kernel<<<cdiv((long long)NN * H1 * 32, T), T, 0, stream>>>(h1, asw1, adw1, as1, ad1, NN, H1);
  edge_max_kernel<<<cdiv(enh1, T), T, 0, stream>>>(ei, as1, ad1, em1, H1);
  edge_expsum_kernel<<<cdiv(enh1, T), T, 0, stream>>>(ei, as1, ad1, em1, dn1, H1);
  edge_aggregate_kernel<<<cdiv(enh1 * 32, T), T, 0, stream>>>(ei, as1, ad1, em1, dn1, h1, acc1, H1);
  finalize_kernel<<<cdiv((long long)NN * OUTC, T), T, 0, stream>>>(acc1, b1, acc1, 1.0f / H1, 1);

  // ---------------- layer 2 ----------------
  fill_u32_kernel<<<cdiv(NN, T), T, 0, stream>>>((uint32_t*)em2, NEG_INF, NN);
  fill_u32_kernel<<<cdiv(NN, T), T, 0, stream>>>((uint32_t*)dn2, 0u, NN);
  fill_u32_kernel<<<cdiv((long long)NN * OUTC, T), T, 0, stream>>>((uint32_t*)out, 0u, (long long)NN * OUTC);

  {  // h2 = elu(h_mid) @ W2  (acc1 holds h_mid)
    long long waves = ((long long)NN / 16) * (OUTC / 64);
    gemm_wmma_f32_kernel<<<cdiv(waves * 32, T), T, 0, stream>>>(acc1, W2, h2, NN, OUTC, OUTC);
  }
  att_scores_kernel<<<cdiv((long long)NN * 32, T), T, 0, stream>>>(h2, asw2, adw2, as2, ad2, NN, 1);
  edge_max_kernel<<<cdiv(enh2, T), T, 0, stream>>>(ei, as2, ad2, em2, 1);
  edge_expsum_kernel<<<cdiv(enh2, T), T, 0, stream>>>(ei, as2, ad2, em2, dn2, 1);
  edge_aggregate_kernel<<<cdiv(enh2 * 32, T), T, 0, stream>>>(ei, as2, ad2, em2, dn2, h2, out, 1);
  finalize_kernel<<<cdiv((long long)NN * OUTC, T), T, 0, stream>>>(out, b2, out, 1.0f, 0);
}